// SamplerQLSTM_65481071405373
// MI455X (gfx1250) — compile-verified
//
#include <hip/hip_runtime.h>
#include <hip/hip_bf16.h>
#include <math.h>

// Problem constants (from reference setup_inputs)
#define T_STEPS 256
#define B_ROWS  128
#define D_DIM   1024
#define H_DIM   1024

typedef float v2f __attribute__((ext_vector_type(2)));
typedef float v8f __attribute__((ext_vector_type(8)));

// ---- workspace layout (float offsets) ----
#define WS_WQI    0        // 16*1024  = Wq@Wi padded to 16 N-rows (rows 4..15 zero)
#define WS_WHSUM  16384    // 1024     = rowsum(Wh)
#define WS_QB     17408    // 4        = Wq@(bi+bh)+bq
#define WS_WQH    17412    // 4        = Wq@whsum
#define WS_CBUF   17424    // 128      = final cbar per row
#define WS_XQ     18432    // 32768*4  = X @ Wqi^T   [T*B][4]
#define WS_HBUF   149504   // 256*128  = hbar[t][b]

// CNOT-chain permutation (precomputed from the reference's _build_tables)
__device__ __constant__ int d_perm[16] =
    {0, 13, 3, 14, 6, 11, 5, 8, 12, 1, 15, 2, 10, 7, 9, 4};

// ---------------------------------------------------------------------------
// prep1: whsum[h] = sum_k Wh[h][k]
__global__ void k_prep1(const float* __restrict__ Wh, float* __restrict__ ws) {
    int h = blockIdx.x * blockDim.x + threadIdx.x;
    if (h >= H_DIM) return;
    const float* row = Wh + (size_t)h * H_DIM;
    float s = 0.f;
    for (int k = 0; k < H_DIM; ++k) s += row[k];
    ws[WS_WHSUM + h] = s;
}

// prep2: Wqi[q][d] = sum_h Wq[q][h] * Wi[h][d], stored padded to 16 rows
// (rows 4..15 zeroed so the WMMA B-fragment loads need no predication).
__global__ void k_prep2(const float* __restrict__ Wi, const float* __restrict__ Wq,
                        float* __restrict__ ws) {
    int d = blockIdx.x * blockDim.x + threadIdx.x;
    if (d >= D_DIM) return;
    float a0 = 0.f, a1 = 0.f, a2 = 0.f, a3 = 0.f;
    for (int h = 0; h < H_DIM; ++h) {
        float w = Wi[(size_t)h * D_DIM + d];
        a0 += Wq[0 * H_DIM + h] * w;
        a1 += Wq[1 * H_DIM + h] * w;
        a2 += Wq[2 * H_DIM + h] * w;
        a3 += Wq[3 * H_DIM + h] * w;
    }
    ws[WS_WQI + 0 * D_DIM + d] = a0;
    ws[WS_WQI + 1 * D_DIM + d] = a1;
    ws[WS_WQI + 2 * D_DIM + d] = a2;
    ws[WS_WQI + 3 * D_DIM + d] = a3;
    #pragma unroll
    for (int q = 4; q < 16; ++q)
        ws[WS_WQI + q * D_DIM + d] = 0.f;
}

// prep3: qb[q] = Wq[q]·(bi+bh) + bq[q];  wqh[q] = Wq[q]·whsum   (1 block, 128 thr)
__global__ void k_prep3(const float* __restrict__ Wq, const float* __restrict__ bi,
                        const float* __restrict__ bh, const float* __restrict__ bq,
                        float* __restrict__ ws) {
    int q    = threadIdx.x >> 5;
    int lane = threadIdx.x & 31;
    float aq = 0.f, aw = 0.f;
    for (int h = lane; h < H_DIM; h += 32) {
        float w = Wq[q * H_DIM + h];
        aq += w * (bi[h] + bh[h]);
        aw += w * ws[WS_WHSUM + h];
    }
    for (int off = 16; off > 0; off >>= 1) {
        aq += __shfl_down(aq, off, 32);
        aw += __shfl_down(aw, off, 32);
    }
    if (lane == 0) {
        ws[WS_QB  + q] = aq + bq[q];
        ws[WS_WQH + q] = aw;
    }
}

// ---------------------------------------------------------------------------
// Xq = X @ Wqi^T  (M=32768, K=1024, N=4 padded to 16) via V_WMMA_F32_16X16X4_F32.
// 256 blocks x 256 threads = 2048 waves; one 16-row M-tile per wave.
// B is pre-padded with zero rows -> fully uniform control flow in the K-loop.
__global__ void k_xq(const float* __restrict__ X, const float* __restrict__ ws_r,
                     float* __restrict__ ws_w) {
    const float* Wqi = ws_r + WS_WQI;
    float*       Xq  = ws_w + WS_XQ;

    int lane  = threadIdx.x & 31;
    int wave  = (blockIdx.x * blockDim.x + threadIdx.x) >> 5;
    int m0    = wave * 16;                 // first row of this wave's M-tile
    int mrow  = m0 + (lane & 15);          // A-matrix row owned by this lane
    int khalf = lane >> 4;                 // 0: K{0,1}, 1: K{2,3} within a k-block
    int n     = lane & 15;                 // B/D column owned by this lane

    const float* xrow = X   + (size_t)mrow * D_DIM + 2 * khalf;
    const float* brow = Wqi + (size_t)n    * D_DIM + 2 * khalf;

    v8f acc = {};
    #pragma unroll 8
    for (int kb = 0; kb < D_DIM; kb += 4) {
        v2f a, b;
        // 32-bit A 16x4 layout: VGPR0 -> K = 2*khalf, VGPR1 -> K = 2*khalf+1
        a.x = xrow[kb];
        a.y = xrow[kb + 1];
        // 32-bit B 4x16 layout mirrors C/D halving: VGPR0 -> K = 2*khalf, VGPR1 -> +1
        b.x = brow[kb];
        b.y = brow[kb + 1];
        acc = __builtin_amdgcn_wmma_f32_16x16x4_f32(
            false, a, false, b, (short)0, acc, false, false);
    }

    // D 16x16 f32: VGPR r holds row (r + 8*khalf), column n = lane&15
    if (n < 4) {
        #pragma unroll
        for (int r = 0; r < 8; ++r) {
            int m = m0 + r + 8 * khalf;
            Xq[(size_t)m * 4 + n] = acc[r];
        }
    }
}

// ---------------------------------------------------------------------------
// Sequential scan: one thread per batch row; 256 steps of the collapsed recurrence.
__global__ void k_scan(const float* __restrict__ theta, float* __restrict__ ws) {
    int b = threadIdx.x;
    if (b >= B_ROWS) return;

    const float4* Xq   = (const float4*)(ws + WS_XQ);
    float*        hbuf = ws + WS_HBUF;

    float qb0 = ws[WS_QB + 0],  qb1 = ws[WS_QB + 1];
    float qb2 = ws[WS_QB + 2],  qb3 = ws[WS_QB + 3];
    float wq0 = ws[WS_WQH + 0], wq1 = ws[WS_WQH + 1];
    float wq2 = ws[WS_WQH + 2], wq3 = ws[WS_WQH + 3];
    float th0 = theta[0], th1 = theta[1], th2 = theta[2], th3 = theta[3];

    float hbar = 0.f, cbar = 0.f;

    for (int t = 0; t < T_STEPS; ++t) {
        float4 xq = Xq[t * B_ROWS + b];
        float qin0 = xq.x + qb0 + hbar * wq0;
        float qin1 = xq.y + qb1 + hbar * wq1;
        float qin2 = xq.z + qb2 + hbar * wq2;
        float qin3 = xq.w + qb3 + hbar * wq3;

        // RX angles add; amplitudes are cos (real) / -i*sin (imag) -> probs use cos^2/sin^2
        float c2[4], s2[4];
        {
            float hl, cc, ss;
            hl = 0.5f * (qin0 + th0); cc = cosf(hl); ss = sinf(hl); c2[0] = cc*cc; s2[0] = ss*ss;
            hl = 0.5f * (qin1 + th1); cc = cosf(hl); ss = sinf(hl); c2[1] = cc*cc; s2[1] = ss*ss;
            hl = 0.5f * (qin2 + th2); cc = cosf(hl); ss = sinf(hl); c2[2] = cc*cc; s2[2] = ss*ss;
            hl = 0.5f * (qin3 + th3); cc = cosf(hl); ss = sinf(hl); c2[3] = cc*cc; s2[3] = ss*ss;
        }

        // exps[w] = sum_j probs[perm[j]] * (1 - 2*bit_w(j))
        float ez0 = 0.f, ez1 = 0.f, ez2 = 0.f, ez3 = 0.f;
        #pragma unroll
        for (int j = 0; j < 16; ++j) {
            int p = d_perm[j];
            float pr = ((p >> 3) & 1 ? s2[0] : c2[0]) *
                       ((p >> 2) & 1 ? s2[1] : c2[1]) *
                       ((p >> 1) & 1 ? s2[2] : c2[2]) *
                       ((p      ) & 1 ? s2[3] : c2[3]);
            ez0 += (j & 8) ? -pr : pr;
            ez1 += (j & 4) ? -pr : pr;
            ez2 += (j & 2) ? -pr : pr;
            ez3 += (j & 1) ? -pr : pr;
        }

        // stable softmax
        float mx = fmaxf(fmaxf(ez0, ez1), fmaxf(ez2, ez3));
        float e0 = expf(ez0 - mx), e1 = expf(ez1 - mx);
        float e2 = expf(ez2 - mx), e3 = expf(ez3 - mx);
        float inv = 1.f / (e0 + e1 + e2 + e3);
        float gf = e0 * inv, gi = e1 * inv, gg = e2 * inv, go = e3 * inv;

        cbar = gf * cbar + gi * tanhf(gg);
        hbar = go * tanhf(cbar);

        hbuf[t * B_ROWS + b] = hbar;
    }
    ws[WS_CBUF + b] = cbar;
}

// ---------------------------------------------------------------------------
// Broadcast hbar/cbar into d_out = concat(ys[T,B,H], hx[B,H], cx[B,H]) as float4s.
__global__ void k_bcast(const float* __restrict__ ws, float* __restrict__ out, int n4) {
    int i4 = blockIdx.x * blockDim.x + threadIdx.x;
    if (i4 >= n4) return;
    const float* hbuf = ws + WS_HBUF;
    const float* cbuf = ws + WS_CBUF;
    const int YS4 = T_STEPS * B_ROWS * (H_DIM / 4);   // 8,388,608
    const int HX4 = B_ROWS * (H_DIM / 4);             // 32,768

    float v;
    if (i4 < YS4) {
        int flat = i4 << 2;                 // element index in ys
        int t = flat >> 17;                 // / (B*H) = / 131072
        int b = (flat >> 10) & (B_ROWS - 1);
        v = hbuf[t * B_ROWS + b];
    } else if (i4 < YS4 + HX4) {
        int flat = (i4 - YS4) << 2;
        int b = (flat >> 10) & (B_ROWS - 1);
        v = hbuf[(T_STEPS - 1) * B_ROWS + b];
    } else {
        int flat = (i4 - YS4 - HX4) << 2;
        int b = (flat >> 10) & (B_ROWS - 1);
        v = cbuf[b];
    }
    ((float4*)out)[i4] = make_float4(v, v, v, v);
}

// ---------------------------------------------------------------------------
extern "C" void kernel_launch(void* const* d_in, const int* in_sizes, int n_in,
                              void* d_out, int out_size, void* d_ws, size_t ws_size,
                              hipStream_t stream) {
    const float* X     = (const float*)d_in[0];  // [256,128,1024]
    const float* Wi    = (const float*)d_in[1];  // [1024,1024]
    const float* bi    = (const float*)d_in[2];  // [1024]
    const float* Wh    = (const float*)d_in[3];  // [1024,1024]
    const float* bh    = (const float*)d_in[4];  // [1024]
    const float* Wq    = (const float*)d_in[5];  // [4,1024]
    const float* bq    = (const float*)d_in[6];  // [4]
    const float* theta = (const float*)d_in[7];  // [4]
    float* out = (float*)d_out;
    float* ws  = (float*)d_ws;

    k_prep1<<<(H_DIM + 255) / 256, 256, 0, stream>>>(Wh, ws);
    k_prep2<<<(D_DIM + 255) / 256, 256, 0, stream>>>(Wi, Wq, ws);
    k_prep3<<<1, 128, 0, stream>>>(Wq, bi, bh, bq, ws);

    // 2048 M-tiles of 16 rows; 8 waves per 256-thread block -> 256 blocks
    k_xq<<<256, 256, 0, stream>>>(X, ws, ws);

    k_scan<<<1, 128, 0, stream>>>(theta, ws);

    int n4 = out_size / 4;                       // 8,454,144 float4s
    k_bcast<<<(n4 + 255) / 256, 256, 0, stream>>>(ws, out, n4);
}